// GraphSAGEWithSampling_41248865911074
// MI455X (gfx1250) — compile-verified
//
#include <hip/hip_runtime.h>
#include <hip/hip_bf16.h>

#define NNODES 100000
#define NEDGES 1600000
#define FIN    128
#define HID    64
#define NCLS   7

typedef float v2f __attribute__((ext_vector_type(2)));
typedef float v8f __attribute__((ext_vector_type(8)));

// ---------------------------------------------------------------------------
// Pass A: in-degree per destination node (one float atomic per edge)
// ---------------------------------------------------------------------------
__global__ void sage_degree(const int* __restrict__ dst, float* __restrict__ cnt) {
    int e = blockIdx.x * blockDim.x + threadIdx.x;
    if (e < NEDGES) atomicAdd(&cnt[dst[e]], 1.0f);
}

// ---------------------------------------------------------------------------
// Pass B: scatter-sum of x[src] into agg1[dst]  (32 lanes per edge, float4)
// ---------------------------------------------------------------------------
__global__ void sage_scatter1(const float* __restrict__ x,
                              const int* __restrict__ src,
                              const int* __restrict__ dst,
                              float* __restrict__ agg1) {
    int gid = blockIdx.x * blockDim.x + threadIdx.x;
    int e = gid >> 5;            // edge
    int l = gid & 31;            // 4-feature chunk within 128
    if (e >= NEDGES) return;
    int s = src[e];
    int d = dst[e];
    float4 v = ((const float4*)(x + (size_t)s * FIN))[l];
    float* o = agg1 + (size_t)d * FIN + l * 4;
    atomicAdd(o + 0, v.x);
    atomicAdd(o + 1, v.y);
    atomicAdd(o + 2, v.z);
    atomicAdd(o + 3, v.w);
}

// ---------------------------------------------------------------------------
// Layer 1: h = relu(mean(agg1) @ W1l + x @ W1r + b1) via V_WMMA_F32_16X16X4_F32
// One wave per 16x16 output tile. 6250 row-tiles x 4 col-tiles = 25000 waves.
// A fragment layout (32-bit A 16x4): lane = (k>>1)*16 + m, vgpr = k&1
// C/D layout: vgpr j -> M = j + 8*(lane>=16), N = lane&15
// ---------------------------------------------------------------------------
__global__ void sage_layer1_wmma(const float* __restrict__ agg1,
                                 const float* __restrict__ x,
                                 const float* __restrict__ cnt,
                                 const float* __restrict__ W1l,
                                 const float* __restrict__ W1r,
                                 const float* __restrict__ b1,
                                 float* __restrict__ h) {
    const int wavesPerBlock = 8;                       // 256 threads / wave32
    int wid   = blockIdx.x * wavesPerBlock + (threadIdx.x >> 5);
    int lane  = threadIdx.x & 31;
    int rowTile = wid >> 2;                            // N/16 tiles
    int colTile = wid & 3;                             // H/16 tiles

    int m     = lane & 15;                             // M (A) / N (B,C,D)
    int khalf = lane >> 4;                             // which K-pair this lane holds

    int row = rowTile * 16 + m;
    float rc = 1.0f / fmaxf(cnt[row], 1.0f);           // mean = sum * 1/deg

    const float* A1 = agg1 + (size_t)row * FIN;        // aggregated neighbors
    const float* A2 = x    + (size_t)row * FIN;        // root features
    const int    nc = colTile * 16 + m;                // output column

    v8f acc = {0.f, 0.f, 0.f, 0.f, 0.f, 0.f, 0.f, 0.f};

    #pragma unroll
    for (int kb = 0; kb < FIN; kb += 4) {
        int k0 = kb + 2 * khalf;

        v2f a, b;
        // agg1 @ W1l  (mean folded into A load)
        a.x = A1[k0] * rc;  a.y = A1[k0 + 1] * rc;
        b.x = W1l[(size_t)k0 * HID + nc];
        b.y = W1l[(size_t)(k0 + 1) * HID + nc];
        acc = __builtin_amdgcn_wmma_f32_16x16x4_f32(false, a, false, b,
                                                    (short)0, acc, false, false);
        // x @ W1r
        a.x = A2[k0];       a.y = A2[k0 + 1];
        b.x = W1r[(size_t)k0 * HID + nc];
        b.y = W1r[(size_t)(k0 + 1) * HID + nc];
        acc = __builtin_amdgcn_wmma_f32_16x16x4_f32(false, a, false, b,
                                                    (short)0, acc, false, false);
    }

    float bias = b1[nc];
    #pragma unroll
    for (int j = 0; j < 8; ++j) {
        int M = j + 8 * khalf;
        float v = acc[j] + bias;
        v = fmaxf(v, 0.0f);                            // ReLU
        h[(size_t)(rowTile * 16 + M) * HID + nc] = v;
    }
}

// ---------------------------------------------------------------------------
// Pass C: scatter-sum of h[src] into agg2[dst]  (16 lanes per edge, float4)
// ---------------------------------------------------------------------------
__global__ void sage_scatter2(const float* __restrict__ h,
                              const int* __restrict__ src,
                              const int* __restrict__ dst,
                              float* __restrict__ agg2) {
    int gid = blockIdx.x * blockDim.x + threadIdx.x;
    int e = gid >> 4;
    int l = gid & 15;
    if (e >= NEDGES) return;
    int s = src[e];
    int d = dst[e];
    float4 v = ((const float4*)(h + (size_t)s * HID))[l];
    float* o = agg2 + (size_t)d * HID + l * 4;
    atomicAdd(o + 0, v.x);
    atomicAdd(o + 1, v.y);
    atomicAdd(o + 2, v.z);
    atomicAdd(o + 3, v.w);
}

// ---------------------------------------------------------------------------
// Layer 2 + log_softmax: out = log_softmax(mean(agg2)@W2l + h@W2r + b2)
// C=7 -> tiny GEMM, per-node VALU loop (~900 FLOP/node)
// ---------------------------------------------------------------------------
__global__ void sage_layer2(const float* __restrict__ agg2,
                            const float* __restrict__ h,
                            const float* __restrict__ cnt,
                            const float* __restrict__ W2l,
                            const float* __restrict__ W2r,
                            const float* __restrict__ b2,
                            float* __restrict__ out) {
    int i = blockIdx.x * blockDim.x + threadIdx.x;
    if (i >= NNODES) return;
    float rc = 1.0f / fmaxf(cnt[i], 1.0f);

    float o[NCLS];
    #pragma unroll
    for (int c = 0; c < NCLS; ++c) o[c] = b2[c];

    const float* a2 = agg2 + (size_t)i * HID;
    const float* hv = h    + (size_t)i * HID;
    for (int k = 0; k < HID; ++k) {
        float a  = a2[k] * rc;
        float hh = hv[k];
        #pragma unroll
        for (int c = 0; c < NCLS; ++c)
            o[c] += a * W2l[k * NCLS + c] + hh * W2r[k * NCLS + c];
    }

    float mx = o[0];
    #pragma unroll
    for (int c = 1; c < NCLS; ++c) mx = fmaxf(mx, o[c]);
    float s = 0.f;
    #pragma unroll
    for (int c = 0; c < NCLS; ++c) s += expf(o[c] - mx);
    float ls = logf(s);
    #pragma unroll
    for (int c = 0; c < NCLS; ++c)
        out[(size_t)i * NCLS + c] = o[c] - mx - ls;
}

// ---------------------------------------------------------------------------
extern "C" void kernel_launch(void* const* d_in, const int* in_sizes, int n_in,
                              void* d_out, int out_size, void* d_ws, size_t ws_size,
                              hipStream_t stream) {
    const float* x    = (const float*)d_in[0];
    const int*   ei   = (const int*)d_in[1];
    const float* W1l  = (const float*)d_in[2];
    const float* W1r  = (const float*)d_in[3];
    const float* b1   = (const float*)d_in[4];
    const float* W2l  = (const float*)d_in[5];
    const float* W2r  = (const float*)d_in[6];
    const float* b2   = (const float*)d_in[7];
    float* out = (float*)d_out;

    const int* src = ei;            // edge_index[0]
    const int* dst = ei + NEDGES;   // edge_index[1]

    // Workspace layout
    char* ws = (char*)d_ws;
    float* agg1 = (float*)ws;                                   // N * FIN
    float* hbuf = (float*)(ws + (size_t)NNODES * FIN * 4);      // N * HID
    float* agg2 = (float*)(ws + (size_t)NNODES * (FIN + HID) * 4);        // N * HID
    float* cnt  = (float*)(ws + (size_t)NNODES * (FIN + 2 * HID) * 4);    // N

    // Zero accumulators every call (harness does not re-poison between replays)
    hipMemsetAsync(agg1, 0, (size_t)NNODES * FIN * 4, stream);
    hipMemsetAsync(agg2, 0, (size_t)NNODES * HID * 4, stream);
    hipMemsetAsync(cnt,  0, (size_t)NNODES * 4, stream);

    // Degree
    sage_degree<<<(NEDGES + 255) / 256, 256, 0, stream>>>(dst, cnt);
    // Scatter-sum layer 1
    sage_scatter1<<<(NEDGES * 32) / 256, 256, 0, stream>>>(x, src, dst, agg1);
    // Layer 1 GEMM (WMMA f32 16x16x4): 25000 waves, 8 waves/block
    sage_layer1_wmma<<<(NNODES / 16) * (HID / 16) / 8, 256, 0, stream>>>(
        agg1, x, cnt, W1l, W1r, b1, hbuf);
    // Scatter-sum layer 2
    sage_scatter2<<<(NEDGES * 16) / 256, 256, 0, stream>>>(hbuf, src, dst, agg2);
    // Layer 2 + log_softmax
    sage_layer2<<<(NNODES + 255) / 256, 256, 0, stream>>>(
        agg2, hbuf, cnt, W2l, W2r, b2, out);
}